// DecoderRNN_60395830116365
// MI455X (gfx1250) — compile-verified
//
#include <hip/hip_runtime.h>
#include <hip/hip_bf16.h>

typedef __attribute__((ext_vector_type(2))) float v2f;
typedef __attribute__((ext_vector_type(4))) float v4f;
typedef __attribute__((ext_vector_type(8))) float v8f;

#define H_DIM 2048
#define V_DIM 50257
#define M_DIM 64

// ---------------------------------------------------------------------------
// K1: attention scores over M=64 rows of attn_w [M, 2H], then softmax.
// One block of 1024 threads; 16 lanes per row; shfl reduce within 16-groups.
// ---------------------------------------------------------------------------
__global__ void __launch_bounds__(1024)
attn_kernel(const int* __restrict__ tok, const float* __restrict__ hidden,
            const float* __restrict__ emb, const float* __restrict__ attn_w,
            const float* __restrict__ attn_b, float* __restrict__ aw_ws,
            float* __restrict__ aw_out) {
  __shared__ float sm[M_DIM];
  const int t = tok[0];
  const float* e = emb + (size_t)t * H_DIM;
  const int g = threadIdx.x >> 4;   // row 0..63
  const int l = threadIdx.x & 15;
  const float* wr = attn_w + (size_t)g * (2 * H_DIM);
  float p = 0.f;
  for (int i = l; i < 2 * H_DIM; i += 16) {
    float c = (i < H_DIM) ? e[i] : hidden[i - H_DIM];
    p += wr[i] * c;
  }
  // reduce across the 16 lanes of this group (xor <= 8 stays in half-wave)
  for (int off = 8; off > 0; off >>= 1) p += __shfl_xor(p, off, 32);
  if (l == 0) sm[g] = p + attn_b[g];
  __syncthreads();
  if (threadIdx.x == 0) {
    float mx = sm[0];
    for (int i = 1; i < M_DIM; i++) mx = fmaxf(mx, sm[i]);
    float s = 0.f;
    for (int i = 0; i < M_DIM; i++) { float v = expf(sm[i] - mx); sm[i] = v; s += v; }
    float inv = 1.f / s;
    for (int i = 0; i < M_DIM; i++) {
      float v = sm[i] * inv;
      aw_ws[i] = v;
      aw_out[i] = v;
    }
  }
}

// ---------------------------------------------------------------------------
// K2: xcat[0..H) = embedded; xcat[H..2H) = attn_weights @ encoder_outputs
// ---------------------------------------------------------------------------
__global__ void __launch_bounds__(256)
xcat_kernel(const int* __restrict__ tok, const float* __restrict__ emb,
            const float* __restrict__ aw, const float* __restrict__ enc,
            float* __restrict__ xcat) {
  int i = blockIdx.x * 256 + threadIdx.x;
  if (i >= 2 * H_DIM) return;
  if (i < H_DIM) {
    xcat[i] = emb[(size_t)tok[0] * H_DIM + i];
  } else {
    int h = i - H_DIM;
    float a = 0.f;
    for (int m = 0; m < M_DIM; m++) a += aw[m] * enc[(size_t)m * H_DIM + h];
    xcat[i] = a;
  }
}

// ---------------------------------------------------------------------------
// WMMA f32 GEMV: y[0..R) = act(W[R,C] @ v[C] + bias[R])
// One wave per 16 output rows. Input vector staged in LDS and broadcast into
// all 16 B-columns. K processed 16 at a time via 4x v_wmma_f32_16x16x4_f32.
// Since B is a broadcast, K-slot assignment is free: lanes 0-15 carry columns
// kk+0..3 / kk+8..11 of their row, lanes 16-31 carry kk+4..7 / kk+12..15,
// so every lane does aligned b128 loads.
// ---------------------------------------------------------------------------
__global__ void __launch_bounds__(256)
gemv16_wmma(const float* __restrict__ W, const float* __restrict__ v,
            const float* __restrict__ bias, float* __restrict__ y,
            int R, int C, int act) {
  __shared__ float sv[2 * H_DIM];  // up to C = 4096 floats (16 KB)
  for (int i = threadIdx.x; i < C; i += 256) sv[i] = v[i];
  __syncthreads();

  const int wave = threadIdx.x >> 5;
  const int lane = threadIdx.x & 31;
  const int tile = blockIdx.x * 8 + wave;
  const int row0 = tile * 16;
  const int m    = lane & 15;
  const int half = lane >> 4;

  int row = row0 + m;
  if (row > R - 1) row = R - 1;  // clamp (no divergence; EXEC stays all-ones)
  const float* wrow = W + (size_t)row * C + half * 4;

  v8f acc = {0.f, 0.f, 0.f, 0.f, 0.f, 0.f, 0.f, 0.f};
  for (int kk = 0; kk < C; kk += 16) {
    v4f q0 = *(const v4f*)(wrow + kk);       // row cols kk+4h .. kk+4h+3
    v4f q1 = *(const v4f*)(wrow + kk + 8);   // row cols kk+4h+8 .. kk+4h+11
    const int xb = kk + half * 4;
    v4f x0 = *(const v4f*)(sv + xb);
    v4f x1 = *(const v4f*)(sv + xb + 8);

    v2f a, b;
    a.x = q0.x; a.y = q0.y; b.x = x0.x; b.y = x0.y;
    acc = __builtin_amdgcn_wmma_f32_16x16x4_f32(false, a, false, b, (short)0, acc, false, false);
    a.x = q0.z; a.y = q0.w; b.x = x0.z; b.y = x0.w;
    acc = __builtin_amdgcn_wmma_f32_16x16x4_f32(false, a, false, b, (short)0, acc, false, false);
    a.x = q1.x; a.y = q1.y; b.x = x1.x; b.y = x1.y;
    acc = __builtin_amdgcn_wmma_f32_16x16x4_f32(false, a, false, b, (short)0, acc, false, false);
    a.x = q1.z; a.y = q1.w; b.x = x1.z; b.y = x1.w;
    acc = __builtin_amdgcn_wmma_f32_16x16x4_f32(false, a, false, b, (short)0, acc, false, false);
  }

  // Column N=0 of D lives in lane 0 (M=0..7) and lane 16 (M=8..15).
  if ((lane & 15) == 0) {
    const int base = row0 + half * 8;
    #pragma unroll
    for (int i = 0; i < 8; i++) {
      int r = base + i;
      if (r < R) {
        float val = acc[i] + bias[r];
        if (act == 1) val = fmaxf(val, 0.f);
        y[r] = val;
      }
    }
  }
}

// ---------------------------------------------------------------------------
// K6: GRU cell elementwise combine (PyTorch gate order r, z, n)
// ---------------------------------------------------------------------------
__global__ void __launch_bounds__(256)
gru_kernel(const float* __restrict__ gi, const float* __restrict__ gh,
           const float* __restrict__ h0, float* __restrict__ hnew_ws,
           float* __restrict__ hnew_out) {
  int i = blockIdx.x * 256 + threadIdx.x;
  if (i >= H_DIM) return;
  float r = 1.f / (1.f + expf(-(gi[i] + gh[i])));
  float z = 1.f / (1.f + expf(-(gi[H_DIM + i] + gh[H_DIM + i])));
  float n = tanhf(gi[2 * H_DIM + i] + r * gh[2 * H_DIM + i]);
  float h = (1.f - z) * n + z * h0[i];
  hnew_ws[i] = h;
  hnew_out[i] = h;
}

// ---------------------------------------------------------------------------
// K8: single-block max + log-sum-exp over the V logits
// ---------------------------------------------------------------------------
__global__ void __launch_bounds__(1024)
lsm_reduce(const float* __restrict__ logits, float* __restrict__ stats, int n) {
  __shared__ float red[1024];
  float mx = -INFINITY;
  for (int i = threadIdx.x; i < n; i += 1024) mx = fmaxf(mx, logits[i]);
  red[threadIdx.x] = mx;
  __syncthreads();
  for (int s = 512; s > 0; s >>= 1) {
    if (threadIdx.x < s) red[threadIdx.x] = fmaxf(red[threadIdx.x], red[threadIdx.x + s]);
    __syncthreads();
  }
  mx = red[0];
  __syncthreads();
  float sum = 0.f;
  for (int i = threadIdx.x; i < n; i += 1024) sum += expf(logits[i] - mx);
  red[threadIdx.x] = sum;
  __syncthreads();
  for (int s = 512; s > 0; s >>= 1) {
    if (threadIdx.x < s) red[threadIdx.x] += red[threadIdx.x + s];
    __syncthreads();
  }
  if (threadIdx.x == 0) {
    stats[0] = mx;
    stats[1] = logf(red[0]);
  }
}

__global__ void __launch_bounds__(256)
lsm_apply(float* __restrict__ logits, const float* __restrict__ stats, int n) {
  int i = blockIdx.x * 256 + threadIdx.x;
  if (i < n) logits[i] = logits[i] - stats[0] - stats[1];
}

// ---------------------------------------------------------------------------
extern "C" void kernel_launch(void* const* d_in, const int* in_sizes, int n_in,
                              void* d_out, int out_size, void* d_ws, size_t ws_size,
                              hipStream_t stream) {
  const int*   tok    = (const int*)  d_in[0];
  const float* hidden = (const float*)d_in[1];
  const float* enc    = (const float*)d_in[2];
  const float* emb    = (const float*)d_in[3];
  const float* attn_w = (const float*)d_in[4];
  const float* attn_b = (const float*)d_in[5];
  const float* comb_w = (const float*)d_in[6];
  const float* comb_b = (const float*)d_in[7];
  const float* w_ih   = (const float*)d_in[8];
  const float* w_hh   = (const float*)d_in[9];
  const float* b_ih   = (const float*)d_in[10];
  const float* b_hh   = (const float*)d_in[11];
  const float* out_w  = (const float*)d_in[12];
  const float* out_b  = (const float*)d_in[13];

  float* out = (float*)d_out;           // [0,V) logprobs | [V,V+H) h_new | [V+H, +64) attn
  float* ws  = (float*)d_ws;
  float* ws_aw    = ws;                 // 64
  float* ws_xcat  = ws + 64;            // 4096
  float* ws_x     = ws + 64 + 4096;     // 2048
  float* ws_gi    = ws + 6208;          // 6144
  float* ws_gh    = ws + 12352;         // 6144
  float* ws_hnew  = ws + 18496;         // 2048
  float* ws_stats = ws + 20544;         // 2

  // 1) attention weights (also final output slice)
  attn_kernel<<<1, 1024, 0, stream>>>(tok, hidden, emb, attn_w, attn_b,
                                      ws_aw, out + V_DIM + H_DIM);
  // 2) xcat = [embedded ; attn_applied]
  xcat_kernel<<<(2 * H_DIM + 255) / 256, 256, 0, stream>>>(tok, emb, ws_aw, enc, ws_xcat);
  // 3) x = relu(comb_w @ xcat + comb_b)   R=2048, C=4096 -> 128 tiles / 16 blocks
  gemv16_wmma<<<16, 256, 0, stream>>>(comb_w, ws_xcat, comb_b, ws_x, H_DIM, 2 * H_DIM, 1);
  // 4) gi = w_ih @ x + b_ih               R=6144, C=2048 -> 384 tiles / 48 blocks
  gemv16_wmma<<<48, 256, 0, stream>>>(w_ih, ws_x, b_ih, ws_gi, 3 * H_DIM, H_DIM, 0);
  // 5) gh = w_hh @ h0 + b_hh
  gemv16_wmma<<<48, 256, 0, stream>>>(w_hh, hidden, b_hh, ws_gh, 3 * H_DIM, H_DIM, 0);
  // 6) GRU combine -> h_new (ws + output slice)
  gru_kernel<<<(H_DIM + 255) / 256, 256, 0, stream>>>(ws_gi, ws_gh, hidden,
                                                      ws_hnew, out + V_DIM);
  // 7) logits = out_w @ h_new + out_b     R=50257 -> 3142 tiles / 393 blocks
  {
    int tiles  = (V_DIM + 15) / 16;
    int blocks = (tiles + 7) / 8;
    gemv16_wmma<<<blocks, 256, 0, stream>>>(out_w, ws_hnew, out_b, out, V_DIM, H_DIM, 0);
  }
  // 8) log-softmax stats, 9) apply in place
  lsm_reduce<<<1, 1024, 0, stream>>>(out, ws_stats, V_DIM);
  lsm_apply<<<(V_DIM + 255) / 256, 256, 0, stream>>>(out, ws_stats, V_DIM);
}